// FallbackRenderer_69329362092672
// MI455X (gfx1250) — compile-verified
//
#include <hip/hip_runtime.h>

#define W 224
#define HALF_W 112.0f
#define FOCALF 500.0f
#define NPIX (W * W)
#define BATCH 32            // points per wave per WMMA pair
#define ITERS 4             // batches per wave -> 128 points per wave

typedef __attribute__((ext_vector_type(2))) float v2f;
typedef __attribute__((ext_vector_type(8))) float v8f;

// ---------------------------------------------------------------------------
// Pass 0: depth buffer <- +inf bits, output image <- 0
// ---------------------------------------------------------------------------
__global__ void pc_init_kernel(float* __restrict__ out, unsigned* __restrict__ depth) {
  int i = blockIdx.x * blockDim.x + threadIdx.x;
  if (i < 3 * NPIX) out[i] = 0.0f;
  if (i < NPIX)     depth[i] = 0x7F800000u;  // +inf bit pattern
}

// ---------------------------------------------------------------------------
// B fragment for one 32-point batch half (branch-free, always-in-bounds loads):
//   lanes 0-15  (sel=0): VGPR0 = x - cx, VGPR1 = y - cy   (K0, K1)
//   lanes 16-31 (sel=1): VGPR0 = z - cz, VGPR1 = 0        (K2, K3)
// ---------------------------------------------------------------------------
__device__ __forceinline__ v2f pc_make_b(const float* __restrict__ pts, int q, int n,
                                         float c0, float c1, int sel) {
  int qc = (q < n) ? q : (n - 1);
  float live = (q < n) ? 1.0f : 0.0f;
  float t0 = pts[qc * 3 + (sel ? 2 : 0)] - c0;   // x or z
  float t1 = pts[qc * 3 + 1] - c1;               // y (always in-bounds)
  v2f b;
  b.x = t0 * live;
  b.y = sel ? 0.0f : t1 * live;
  return b;
}

// Projection exactly matching reference evaluation order:
//   x = (xc*FOCAL)/z + W//2 ; bounds-check on floats ; trunc to int
__device__ __forceinline__ int pc_project(float xc, float yc, float zc) {
  if (!(zc > 0.1f)) return -1;
  float x = (xc * FOCALF) / zc + HALF_W;
  float y = (yc * FOCALF) / zc + HALF_W;
  if (!(x >= 0.0f && x < (float)W && y >= 0.0f && y < (float)W)) return -1;
  return (int)y * W + (int)x;
}

// ---------------------------------------------------------------------------
// Shared per-wave body: dual-WMMA transform of 32 points.
//   A (16x4): rows 0-2 AND 8-10 hold R (duplicated); else zero.
//   D1 rows 0-2  -> lanes 0-15,  VGPRs 0-2  (points base+0..15)
//   D2 rows 8-10 -> lanes 16-31, VGPRs 0-2  (points base+16..31)
// EXEC is all-ones at both WMMAs (no divergence before them).
// ---------------------------------------------------------------------------
#define PC_WAVE_SETUP()                                                        \
  const int lane = threadIdx.x & 31;                                           \
  const int wave = (blockIdx.x * blockDim.x + threadIdx.x) >> 5;               \
  const int sel  = lane >> 4;                                                  \
  const int j    = lane & 15;                                                  \
  const int base0 = wave * (BATCH * ITERS);                                    \
  if (base0 >= n) return;                                                      \
  const int rr  = (lane & 15) & 7;                                             \
  const bool useA = (rr < 3);                                                  \
  const int rc  = useA ? rr : 2;                                               \
  float ra0 = rot[rc * 3 + (sel ? 2 : 0)];                                     \
  float ra1 = rot[rc * 3 + 1];                                                 \
  v2f a;                                                                       \
  a.x = useA ? ra0 : 0.0f;                                                     \
  a.y = (useA && !sel) ? ra1 : 0.0f;                                           \
  const float c0 = campos[sel ? 2 : 0];                                        \
  const float c1 = campos[1];

#define PC_TRANSFORM32(base)                                                   \
  __builtin_prefetch(pts + (size_t)((base) + BATCH) * 3 + (size_t)lane * 3,    \
                     0, 1);                                                    \
  v2f b1 = pc_make_b(pts, (base) + j,      n, c0, c1, sel);                    \
  v2f b2 = pc_make_b(pts, (base) + 16 + j, n, c0, c1, sel);                    \
  v8f z8 = {};                                                                 \
  v8f d1 = __builtin_amdgcn_wmma_f32_16x16x4_f32(false, a, false, b1,          \
                                                 (short)0, z8, false, false);  \
  v8f d2 = __builtin_amdgcn_wmma_f32_16x16x4_f32(false, a, false, b2,          \
                                                 (short)0, z8, false, false);  \
  float xc = sel ? d2[0] : d1[0];                                              \
  float yc = sel ? d2[1] : d1[1];                                              \
  float zc = sel ? d2[2] : d1[2];

// ---------------------------------------------------------------------------
// Pass 1: atomic min of depth bits (positive floats order as u32)
// ---------------------------------------------------------------------------
__global__ void pc_depth_kernel(const float* __restrict__ pts,
                                const float* __restrict__ campos,
                                const float* __restrict__ rot,
                                unsigned* __restrict__ depth, int n) {
  PC_WAVE_SETUP();
  for (int it = 0; it < ITERS; ++it) {
    int base = base0 + it * BATCH;            // wave-uniform
    if (base >= n) break;
    PC_TRANSFORM32(base);
    int p = base + lane;
    if (p < n) {
      int pid = pc_project(xc, yc, zc);
      if (pid >= 0)
        atomicMin(&depth[pid], __float_as_uint(zc));  // global_atomic_min_u32
    }
  }
}

// ---------------------------------------------------------------------------
// Pass 2: winners (bit-exact depth match) write their color, planar CHW
// ---------------------------------------------------------------------------
__global__ void pc_resolve_kernel(const float* __restrict__ pts,
                                  const float* __restrict__ campos,
                                  const float* __restrict__ rot,
                                  const float* __restrict__ colors,
                                  const unsigned* __restrict__ depth,
                                  float* __restrict__ out, int n) {
  PC_WAVE_SETUP();
  for (int it = 0; it < ITERS; ++it) {
    int base = base0 + it * BATCH;
    if (base >= n) break;
    PC_TRANSFORM32(base);
    int p = base + lane;
    if (p < n) {
      int pid = pc_project(xc, yc, zc);
      if (pid >= 0 && __float_as_uint(zc) == depth[pid]) {
        out[0 * NPIX + pid] = colors[p * 3 + 0];
        out[1 * NPIX + pid] = colors[p * 3 + 1];
        out[2 * NPIX + pid] = colors[p * 3 + 2];
      }
    }
  }
}

// ---------------------------------------------------------------------------
extern "C" void kernel_launch(void* const* d_in, const int* in_sizes, int n_in,
                              void* d_out, int out_size, void* d_ws, size_t ws_size,
                              hipStream_t stream) {
  const float* pts    = (const float*)d_in[0];   // (N,3)
  const float* campos = (const float*)d_in[1];   // (3,)
  const float* rot    = (const float*)d_in[2];   // (3,3) row-major
  const float* colors = (const float*)d_in[3];   // (N,3)
  float*    out   = (float*)d_out;               // (3,224,224)
  unsigned* depth = (unsigned*)d_ws;             // 224*224 u32 (~200 KB, L2-resident)

  int n = in_sizes[0] / 3;

  pc_init_kernel<<<(3 * NPIX + 255) / 256, 256, 0, stream>>>(out, depth);

  int ptsPerWave = BATCH * ITERS;                         // 128
  long waves   = ((long)n + ptsPerWave - 1) / ptsPerWave; // 31250 for N=4M
  long threads = waves * 32;
  int  blocks  = (int)((threads + 255) / 256);

  pc_depth_kernel  <<<blocks, 256, 0, stream>>>(pts, campos, rot, depth, n);
  pc_resolve_kernel<<<blocks, 256, 0, stream>>>(pts, campos, rot, colors, depth, out, n);
}